// GNN_Geo_35296041238972
// MI455X (gfx1250) — compile-verified
//
#include <hip/hip_runtime.h>
#include <hip/hip_bf16.h>

// GCN stack for MI455X (gfx1250, wave32).
// Dense X@W via V_WMMA_F32_16X16X4_F32 (exact fp32 matrix path);
// sparse aggregation via device-built CSR (histogram -> scan -> scatter -> sort)
// and a gather-style SpMM with no atomics in the hot loop.

#define NN 4096
#define NE 131072
#define ETOT (NE + NN)
#define NEG_SLOPE 0.01f

typedef float v2f __attribute__((ext_vector_type(2)));
typedef float v8f __attribute__((ext_vector_type(8)));

// ---------------- graph preprocessing ----------------

__global__ void k_init_deg(int* __restrict__ deg) {
    int i = blockIdx.x * blockDim.x + threadIdx.x;
    if (i < NN) deg[i] = 1;  // self loop contributes 1
}

__global__ void k_hist(const int* __restrict__ col, int* __restrict__ deg) {
    int e = blockIdx.x * blockDim.x + threadIdx.x;
    if (e < NE) atomicAdd(&deg[col[e]], 1);
}

__global__ void k_dinv(const int* __restrict__ deg, float* __restrict__ dinv) {
    int i = blockIdx.x * blockDim.x + threadIdx.x;
    if (i < NN) {
        int d = deg[i];
        dinv[i] = (d > 0) ? rsqrtf((float)d) : 0.0f;
    }
}

// Single-block exclusive scan of 4096 degree counts -> rowptr; also zeroes fill[].
__global__ __launch_bounds__(256) void k_scan(const int* __restrict__ deg,
                                              int* __restrict__ rowptr,
                                              int* __restrict__ fill) {
    __shared__ int sums[256];
    int t = threadIdx.x;
    int base = t * 16;
    int local[16];
    int s = 0;
#pragma unroll
    for (int i = 0; i < 16; ++i) { local[i] = s; s += deg[base + i]; }
    sums[t] = s;
    __syncthreads();
    for (int off = 1; off < 256; off <<= 1) {
        int v = (t >= off) ? sums[t - off] : 0;
        __syncthreads();
        sums[t] += v;
        __syncthreads();
    }
    int prev = (t == 0) ? 0 : sums[t - 1];
#pragma unroll
    for (int i = 0; i < 16; ++i) {
        rowptr[base + i] = prev + local[i];
        fill[base + i] = 0;
    }
    if (t == 255) rowptr[NN] = sums[255];  // == ETOT
}

__global__ void k_scatter(const int* __restrict__ ei, const float* __restrict__ dinv,
                          const int* __restrict__ rowptr, int* __restrict__ fill,
                          int* __restrict__ srcidx, float* __restrict__ wgt) {
    int e = blockIdx.x * blockDim.x + threadIdx.x;
    if (e >= ETOT) return;
    int r, c;
    if (e < NE) { r = ei[e]; c = ei[NE + e]; }   // ei[0,:]=row(src), ei[1,:]=col(dst)
    else        { r = c = e - NE; }              // appended self loops
    int pos = rowptr[c] + atomicAdd(&fill[c], 1);
    srcidx[pos] = r;
    wgt[pos] = dinv[r] * dinv[c];
}

// Per-node insertion sort of adjacency slots by source index -> deterministic
// accumulation order in SpMM regardless of atomic scatter ordering.
__global__ void k_sort(const int* __restrict__ rowptr, int* __restrict__ srcidx,
                       float* __restrict__ wgt) {
    int c = blockIdx.x * blockDim.x + threadIdx.x;
    if (c >= NN) return;
    int beg = rowptr[c], end = rowptr[c + 1];
    for (int i = beg + 1; i < end; ++i) {
        int ks = srcidx[i];
        float kw = wgt[i];
        int j = i - 1;
        while (j >= beg && srcidx[j] > ks) {
            srcidx[j + 1] = srcidx[j];
            wgt[j + 1] = wgt[j];
            --j;
        }
        srcidx[j + 1] = ks;
        wgt[j + 1] = kw;
    }
}

// ---------------- dense GEMM via fp32 WMMA ----------------
// C[4096,4096] = op(A) * B, row-major fp32.
// TRANS_A=false: C = A @ B.   TRANS_A=true: C = A^T @ B (for FA^T @ FB).
// Block tile 128(M) x 128(N), BK=32, 256 threads = 8 wave32.
// Wave grid 4x2; each wave computes a 32x64 region = 2x4 WMMA 16x16 tiles.
// Both A and B tiles are stored fragment-major in LDS (k contiguous per lane),
// so every fragment is a single conflict-free ds_load_b64.

#define BM 128
#define BN 128
#define BK 32
#define LSTR (BK + 4)  // 36: keeps float4 stores 16B-aligned, spreads banks

template <bool TRANS_A>
__global__ __launch_bounds__(256) void k_gemm(const float* __restrict__ A,
                                              const float* __restrict__ B,
                                              float* __restrict__ C) {
    __shared__ float As[BM][LSTR];  // As[m][k]
    __shared__ float Bt[BN][LSTR];  // Bt[n][k]  (transposed tile of B)

    const int tid = threadIdx.x;
    const int wave = tid >> 5;      // 0..7
    const int lane = tid & 31;
    const int wm = wave >> 1;       // 0..3 -> M offset 32*wm
    const int wn = wave & 1;        // 0..1 -> N offset 64*wn
    const int ln = lane & 15;       // M/N index within fragment
    const int lg = lane >> 4;       // half-wave selector (K pair)
    const int m0 = blockIdx.y * BM;
    const int n0 = blockIdx.x * BN;

    v8f acc[2][4] = {};

    for (int k0 = 0; k0 < NN; k0 += BK) {
        // ---- stage A tile: As[m][k] = op(A)[m0+m][k0+k] ----
        if (!TRANS_A) {
#pragma unroll
            for (int p = 0; p < 4; ++p) {
                int idx = tid + p * 256;              // 1024 float4 = 128x32 floats
                int r = idx >> 3, c4 = (idx & 7) << 2;
                *(float4*)&As[r][c4] =
                    *(const float4*)&A[(size_t)(m0 + r) * NN + k0 + c4];
            }
        } else {
#pragma unroll
            for (int p = 0; p < 16; ++p) {
                int idx = tid + p * 256;              // coalesced over m
                int m = idx & 127, k = idx >> 7;
                As[m][k] = A[(size_t)(k0 + k) * NN + m0 + m];
            }
        }
        // ---- stage B tile transposed: Bt[n][k] = B[k0+k][n0+n] ----
        // 4 coalesced scalar reads (n fastest across lanes) + 1 float4 LDS store.
#pragma unroll
        for (int p = 0; p < 4; ++p) {
            int idx = tid + p * 256;                  // 1024 items = 128 n x 8 k-quads
            int n = idx & 127, kq = (idx >> 7) << 2;
            float4 v;
            v.x = B[(size_t)(k0 + kq + 0) * NN + n0 + n];
            v.y = B[(size_t)(k0 + kq + 1) * NN + n0 + n];
            v.z = B[(size_t)(k0 + kq + 2) * NN + n0 + n];
            v.w = B[(size_t)(k0 + kq + 3) * NN + n0 + n];
            *(float4*)&Bt[n][kq] = v;
        }
        __syncthreads();

#pragma unroll
        for (int kk = 0; kk < BK; kk += 4) {
            // A frag 16x4: lanes 0-15 hold K=kk,kk+1; lanes 16-31 hold K=kk+2,kk+3
            v2f a[2];
#pragma unroll
            for (int i = 0; i < 2; ++i)
                a[i] = *(const v2f*)&As[wm * 32 + i * 16 + ln][kk + 2 * lg];
            // B frag 4x16: VGPR0 -> K=kk+2*lg, VGPR1 -> K=kk+2*lg+1, N=ln
            v2f b[4];
#pragma unroll
            for (int j = 0; j < 4; ++j)
                b[j] = *(const v2f*)&Bt[wn * 64 + j * 16 + ln][kk + 2 * lg];
#pragma unroll
            for (int i = 0; i < 2; ++i)
#pragma unroll
                for (int j = 0; j < 4; ++j)
                    acc[i][j] = __builtin_amdgcn_wmma_f32_16x16x4_f32(
                        false, a[i], false, b[j], (short)0, acc[i][j], false, false);
        }
        __syncthreads();
    }

    // C/D layout: VGPR v -> row v + 8*lg, col ln (within 16x16 tile)
#pragma unroll
    for (int i = 0; i < 2; ++i)
#pragma unroll
        for (int j = 0; j < 4; ++j) {
            int mb = m0 + wm * 32 + i * 16 + 8 * lg;
            int nb = n0 + wn * 64 + j * 16 + ln;
#pragma unroll
            for (int v = 0; v < 8; ++v)
                C[(size_t)(mb + v) * NN + nb] = acc[i][j][v];
        }
}

// ---------------- SpMM: out[c,:] = sum_e w[e] * Y[src[e],:]  (+ LeakyReLU) ----
// Y (64MB) resides in the 192MB L2 -> gather traffic stays on-die.
__global__ __launch_bounds__(256) void k_spmm(const float* __restrict__ Y,
                                              const int* __restrict__ rowptr,
                                              const int* __restrict__ srcidx,
                                              const float* __restrict__ wgt,
                                              float* __restrict__ out,
                                              int apply_leaky) {
    const int c = blockIdx.y;
    const int col = blockIdx.x * 256 + threadIdx.x;
    const int beg = rowptr[c], end = rowptr[c + 1];
    float acc = 0.0f;
    for (int e = beg; e < end; ++e) {
        const int s = srcidx[e];
        if (e + 1 < end) {
            // hide L2 latency for the next gathered row
            __builtin_prefetch(&Y[(size_t)srcidx[e + 1] * NN + col], 0, 1);
        }
        acc += wgt[e] * Y[(size_t)s * NN + col];
    }
    if (apply_leaky) acc = (acc > 0.0f) ? acc : NEG_SLOPE * acc;
    out[(size_t)c * NN + col] = acc;
}

// ---------------- orchestration ----------------

extern "C" void kernel_launch(void* const* d_in, const int* in_sizes, int n_in,
                              void* d_out, int out_size, void* d_ws, size_t ws_size,
                              hipStream_t stream) {
    (void)in_sizes; (void)n_in; (void)out_size; (void)ws_size;

    const int*   ei1     = (const int*)d_in[0];
    const int*   ei2     = (const int*)d_in[1];
    const float* feature = (const float*)d_in[2];
    const float* W       = (const float*)d_in[3];

    float* FA = (float*)d_out;                         // [N,N]
    float* FB = FA + (size_t)NN * NN;                  // [N,N]
    float* P  = FB + (size_t)NN * NN;                  // [N,N]

    // workspace carve-out
    uintptr_t base = (uintptr_t)d_ws;
    size_t off = 0;
    auto alloc = [&](size_t bytes) -> void* {
        void* p = (void*)(base + off);
        off += (bytes + 255) & ~(size_t)255;
        return p;
    };
    float* Y  = (float*)alloc((size_t)NN * NN * sizeof(float));
    float* XA = (float*)alloc((size_t)NN * NN * sizeof(float));
    float* XB = (float*)alloc((size_t)NN * NN * sizeof(float));

    int*   deg[2];  float* dinv[2];  int* rowptr[2];  int* fill[2];
    int*   srci[2]; float* wgts[2];
    for (int g = 0; g < 2; ++g) {
        deg[g]    = (int*)  alloc(NN * sizeof(int));
        dinv[g]   = (float*)alloc(NN * sizeof(float));
        rowptr[g] = (int*)  alloc((NN + 1) * sizeof(int));
        fill[g]   = (int*)  alloc(NN * sizeof(int));
        srci[g]   = (int*)  alloc((size_t)ETOT * sizeof(int));
        wgts[g]   = (float*)alloc((size_t)ETOT * sizeof(float));
    }

    // --- build CSR + norm for both graphs ---
    const int* ei[2] = { ei1, ei2 };
    for (int g = 0; g < 2; ++g) {
        k_init_deg<<<NN / 256, 256, 0, stream>>>(deg[g]);
        k_hist<<<NE / 256, 256, 0, stream>>>(ei[g] + NE, deg[g]);
        k_dinv<<<NN / 256, 256, 0, stream>>>(deg[g], dinv[g]);
        k_scan<<<1, 256, 0, stream>>>(deg[g], rowptr[g], fill[g]);
        k_scatter<<<(ETOT + 255) / 256, 256, 0, stream>>>(ei[g], dinv[g], rowptr[g],
                                                          fill[g], srci[g], wgts[g]);
        k_sort<<<NN / 256, 256, 0, stream>>>(rowptr[g], srci[g], wgts[g]);
    }

    // --- 5 stacked GCN convs per graph (last one without activation) ---
    const dim3 gemm_grid(NN / BN, NN / BM);  // (32, 32)
    const dim3 spmm_grid(NN / 256, NN);      // (16, 4096)

    const float* XcurA = feature;
    const float* XcurB = feature;
    for (int l = 0; l < 5; ++l) {
        const bool last = (l == 4);
        float* dstA = last ? FA : XA;
        float* dstB = last ? FB : XB;

        k_gemm<false><<<gemm_grid, 256, 0, stream>>>(XcurA, W, Y);
        k_spmm<<<spmm_grid, 256, 0, stream>>>(Y, rowptr[0], srci[0], wgts[0], dstA,
                                              last ? 0 : 1);
        k_gemm<false><<<gemm_grid, 256, 0, stream>>>(XcurB, W, Y);
        k_spmm<<<spmm_grid, 256, 0, stream>>>(Y, rowptr[1], srci[1], wgts[1], dstB,
                                              last ? 0 : 1);
        XcurA = XA;
        XcurB = XB;
    }

    // --- prediction_aff = FA^T @ FB ---
    k_gemm<true><<<gemm_grid, 256, 0, stream>>>(FA, FB, P);
}